// WindowAttention_60455959658928
// MI455X (gfx1250) — compile-verified
//
#include <hip/hip_runtime.h>

typedef __attribute__((ext_vector_type(2))) float v2f;
typedef __attribute__((ext_vector_type(8))) float v8f;

#define B_TOT 2048
#define NW    64
#define BATCH 32
#define NWIN  49
#define PLEN  20
#define NTOK  69
#define NPAD  80
#define CDIM  128
#define NH    4
#define HD    32
#define QKV3  384
#define SCALE 0.17677669529663687f

#define QP 36   // Q/K/V LDS pitch: 4*9, 9 coprime 16 -> conflict-free columns, 16B rows
#define SP 84   // S LDS pitch: 4*21, 21 coprime 16 -> conflict-free columns
#define AP 132  // A LDS pitch (stride 4 banks, 16B aligned)

#define X_OUT_ELEMS   (B_TOT * NWIN * CDIM)        /* 12845056 */
#define ATTNW_ELEMS   (B_TOT * NH * PLEN * NWIN)   /*  8028160 */
#define PROMPT_ELEMS  (BATCH * PLEN * CDIM)        /*    81920 */

__device__ __forceinline__ v8f wmma4(v2f a, v2f b, v8f c) {
  // D = A(16x4 f32) * B(4x16 f32) + C(16x16 f32)
  return __builtin_amdgcn_wmma_f32_16x16x4_f32(false, a, false, b, (short)0, c,
                                               false, false);
}

// -------------------- Kernel 1: QKV GEMM (per window) --------------------
// qkv[b][n][0:384] = concat(prompts,x)[b][n][:] @ W_qkv + b_qkv
// Each wave: 5(M) x 3(N) register tile block -> 15 WMMAs per 16 operand loads.
__global__ __launch_bounds__(256) void qkv_kernel(
    const float* __restrict__ x, const float* __restrict__ spa,
    const float* __restrict__ Wqkv, const float* __restrict__ bqkv,
    float* __restrict__ qkv_ws) {
  __shared__ float A[NPAD][AP];
  const int b = blockIdx.x, tid = threadIdx.x;
  for (int idx = tid; idx < NPAD * 32; idx += 256) {   // float4 granularity
    int n = idx >> 5, c4 = (idx & 31) << 2;
    float4 v = make_float4(0.f, 0.f, 0.f, 0.f);
    if (n < PLEN)       v = *(const float4*)(spa + ((b / NW) * PLEN + n) * CDIM + c4);
    else if (n < NTOK)  v = *(const float4*)(x + (b * NWIN + (n - PLEN)) * CDIM + c4);
    *(float4*)&A[n][c4] = v;
  }
  __syncthreads();
  const int wave = tid >> 5, lane = tid & 31;
  const int l16 = lane & 15, koff = (lane >> 4) << 1, rsel = (lane >> 4) << 3;

  v8f acc[5][3];
#pragma unroll
  for (int mi = 0; mi < 5; ++mi)
#pragma unroll
    for (int s = 0; s < 3; ++s)
      acc[mi][s] = (v8f){0.f, 0.f, 0.f, 0.f, 0.f, 0.f, 0.f, 0.f};

  int nn[3];
#pragma unroll
  for (int s = 0; s < 3; ++s) nn[s] = (wave + 8 * s) * 16 + l16;

  for (int k = 0; k < CDIM; k += 4) {
    v2f af[5], bf[3];
#pragma unroll
    for (int mi = 0; mi < 5; ++mi)
      af[mi] = (v2f){A[mi * 16 + l16][k + koff], A[mi * 16 + l16][k + koff + 1]};
#pragma unroll
    for (int s = 0; s < 3; ++s)
      bf[s] = (v2f){Wqkv[(k + koff) * QKV3 + nn[s]],
                    Wqkv[(k + koff + 1) * QKV3 + nn[s]]};
#pragma unroll
    for (int mi = 0; mi < 5; ++mi)
#pragma unroll
      for (int s = 0; s < 3; ++s)
        acc[mi][s] = wmma4(af[mi], bf[s], acc[mi][s]);
  }

#pragma unroll
  for (int s = 0; s < 3; ++s) {
    const float bias = bqkv[nn[s]];
#pragma unroll
    for (int mi = 0; mi < 5; ++mi)
#pragma unroll
      for (int r = 0; r < 8; ++r) {
        int row = mi * 16 + r + rsel;
        if (row < NTOK)
          qkv_ws[(b * NTOK + row) * QKV3 + nn[s]] = acc[mi][s][r] + bias;
      }
  }
}

// -------------------- Kernel 2: attention (per window, head) --------------------
__global__ __launch_bounds__(256) void attn_kernel(
    const float* __restrict__ qkv_ws, const float* __restrict__ relb,
    const float* __restrict__ mask, float* __restrict__ attnw_out,
    float* __restrict__ attn_ws) {
  __shared__ float Q[NPAD][QP], K[NPAD][QP], V[NPAD][QP];
  __shared__ float S[NPAD][SP];
  const int b = blockIdx.x >> 2, h = blockIdx.x & 3;
  const int w = b & (NW - 1);
  const int tid = threadIdx.x;
  for (int idx = tid; idx < NPAD * 8; idx += 256) {     // float4 granularity
    int n = idx >> 3, d4 = (idx & 7) << 2;
    float4 q = make_float4(0.f, 0.f, 0.f, 0.f), k = q, v = q;
    if (n < NTOK) {
      const float* p = qkv_ws + (b * NTOK + n) * QKV3 + h * HD + d4;
      q = *(const float4*)p;
      k = *(const float4*)(p + CDIM);
      v = *(const float4*)(p + 2 * CDIM);
    }
    *(float4*)&Q[n][d4] = q;
    *(float4*)&K[n][d4] = k;
    *(float4*)&V[n][d4] = v;
  }
  __syncthreads();
  const int wave = tid >> 5, lane = tid & 31;
  const int l16 = lane & 15, koff = (lane >> 4) << 1, rsel = (lane >> 4) << 3;
  // S = Q @ K^T  (contraction over head-dim 32); padded rows give S==0
  for (int t = wave; t < 25; t += 8) {
    const int mi = t / 5, ni = t % 5;
    v8f acc = {0.f, 0.f, 0.f, 0.f, 0.f, 0.f, 0.f, 0.f};
    for (int d = 0; d < HD; d += 4) {
      v2f af = {Q[mi * 16 + l16][d + koff], Q[mi * 16 + l16][d + koff + 1]};
      v2f bf = {K[ni * 16 + l16][d + koff], K[ni * 16 + l16][d + koff + 1]};
      acc = wmma4(af, bf, acc);
    }
#pragma unroll
    for (int r = 0; r < 8; ++r)
      S[mi * 16 + r + rsel][ni * 16 + l16] = acc[r];
  }
  __syncthreads();
  // raw (pre-scale) prompt->window logits = ori_attn_weight
  for (int idx = tid; idx < PLEN * NWIN; idx += 256) {
    int pi = idx / NWIN, mj = idx % NWIN;
    attnw_out[((b * NH + h) * PLEN + pi) * NWIN + mj] = S[pi][PLEN + mj];
  }
  __syncthreads();
  // scale + relative-position bias + window mask
  for (int idx = tid; idx < NTOK * NTOK; idx += 256) {
    int i = idx / NTOK, j = idx % NTOK;
    float val = S[i][j] * SCALE;
    if (i >= PLEN && j >= PLEN) {
      int ii = i - PLEN, jj = j - PLEN;
      int rel = ((ii / 7) - (jj / 7) + 6) * 13 + ((ii % 7) - (jj % 7) + 6);
      val += relb[rel * NH + h] + mask[(w * NWIN + ii) * NWIN + jj];
    }
    S[i][j] = val;
  }
  __syncthreads();
  // wave-per-row softmax over 69 valid columns (pad cols stay exactly 0)
  for (int r = wave; r < NTOK; r += 8) {
    float v0 = S[r][lane];                                   // lane     < 69 always
    float v1 = S[r][lane + 32];                              // lane+32  < 69 always
    float v2 = (lane < 5) ? S[r][lane + 64] : -3.4e38f;      // lane+64  < 69 iff lane<5
    float mx = fmaxf(fmaxf(v0, v1), v2);
#pragma unroll
    for (int off = 16; off; off >>= 1) mx = fmaxf(mx, __shfl_xor(mx, off, 32));
    float e0 = __expf(v0 - mx), e1 = __expf(v1 - mx);
    float e2 = (lane < 5) ? __expf(v2 - mx) : 0.f;
    float sum = e0 + e1 + e2;
#pragma unroll
    for (int off = 16; off; off >>= 1) sum += __shfl_xor(sum, off, 32);
    float inv = 1.f / sum;
    S[r][lane] = e0 * inv;
    S[r][lane + 32] = e1 * inv;
    if (lane < 5) S[r][lane + 64] = e2 * inv;
  }
  __syncthreads();
  // O = softmax(S) @ V  (contraction over 80, zero padding is exact)
  for (int t = wave; t < 10; t += 8) {
    const int mi = t >> 1, ni = t & 1;
    v8f acc = {0.f, 0.f, 0.f, 0.f, 0.f, 0.f, 0.f, 0.f};
    for (int kk = 0; kk < NPAD; kk += 4) {
      v2f af = {S[mi * 16 + l16][kk + koff], S[mi * 16 + l16][kk + koff + 1]};
      v2f bf = {V[kk + koff][ni * 16 + l16], V[kk + koff + 1][ni * 16 + l16]};
      acc = wmma4(af, bf, acc);
    }
#pragma unroll
    for (int r = 0; r < 8; ++r) {
      int row = mi * 16 + r + rsel;
      if (row < NTOK)
        attn_ws[(b * NTOK + row) * CDIM + h * HD + ni * 16 + l16] = acc[r];
    }
  }
}

// -------------------- Kernel 3: output projection --------------------
// Each wave: one N tile (ni = wave), 5 M accumulators.
__global__ __launch_bounds__(256) void proj_kernel(
    const float* __restrict__ attn_ws, const float* __restrict__ Wproj,
    const float* __restrict__ bproj, float* __restrict__ x_out,
    float* __restrict__ pp_ws) {
  __shared__ float A[NPAD][AP];
  const int b = blockIdx.x, tid = threadIdx.x;
  for (int idx = tid; idx < NPAD * 32; idx += 256) {
    int n = idx >> 5, c4 = (idx & 31) << 2;
    float4 v = make_float4(0.f, 0.f, 0.f, 0.f);
    if (n < NTOK) v = *(const float4*)(attn_ws + (b * NTOK + n) * CDIM + c4);
    *(float4*)&A[n][c4] = v;
  }
  __syncthreads();
  const int wave = tid >> 5, lane = tid & 31;
  const int l16 = lane & 15, koff = (lane >> 4) << 1, rsel = (lane >> 4) << 3;
  const int nn = wave * 16 + l16;

  v8f acc[5];
#pragma unroll
  for (int mi = 0; mi < 5; ++mi)
    acc[mi] = (v8f){0.f, 0.f, 0.f, 0.f, 0.f, 0.f, 0.f, 0.f};

  for (int k = 0; k < CDIM; k += 4) {
    v2f bf = {Wproj[(k + koff) * CDIM + nn], Wproj[(k + koff + 1) * CDIM + nn]};
#pragma unroll
    for (int mi = 0; mi < 5; ++mi) {
      v2f af = {A[mi * 16 + l16][k + koff], A[mi * 16 + l16][k + koff + 1]};
      acc[mi] = wmma4(af, bf, acc[mi]);
    }
  }
  const float bias = bproj[nn];
#pragma unroll
  for (int mi = 0; mi < 5; ++mi)
#pragma unroll
    for (int r = 0; r < 8; ++r) {
      int row = mi * 16 + r + rsel;
      float val = acc[mi][r] + bias;
      if (row < PLEN)       pp_ws[(b * PLEN + row) * CDIM + nn] = val;
      else if (row < NTOK)  x_out[(b * NWIN + (row - PLEN)) * CDIM + nn] = val;
    }
}

// -------------------- Kernel 4: prompt mean over nW (deterministic) --------------------
__global__ __launch_bounds__(256) void prompt_reduce_kernel(
    const float* __restrict__ pp_ws, float* __restrict__ prompts_out) {
  int idx = blockIdx.x * 256 + threadIdx.x;
  if (idx >= PROMPT_ELEMS) return;
  int c = idx & 127;
  int p = (idx >> 7) % PLEN;
  int bb = idx / (PLEN * CDIM);
  float s = 0.f;
  for (int w = 0; w < NW; ++w)
    s += pp_ws[((bb * NW + w) * PLEN + p) * CDIM + c];
  prompts_out[idx] = s * (1.0f / NW);
}

extern "C" void kernel_launch(void* const* d_in, const int* in_sizes, int n_in,
                              void* d_out, int out_size, void* d_ws, size_t ws_size,
                              hipStream_t stream) {
  (void)in_sizes; (void)n_in; (void)out_size; (void)ws_size;
  const float* x     = (const float*)d_in[0];
  const float* spa   = (const float*)d_in[1];
  const float* mask  = (const float*)d_in[2];
  const float* relb  = (const float*)d_in[3];
  const float* Wqkv  = (const float*)d_in[4];
  const float* bqkv  = (const float*)d_in[5];
  const float* Wproj = (const float*)d_in[6];
  const float* bproj = (const float*)d_in[7];

  float* out         = (float*)d_out;
  float* x_out       = out;
  float* attnw_out   = out + X_OUT_ELEMS;
  float* prompts_out = out + X_OUT_ELEMS + ATTNW_ELEMS;

  float* qkv_ws  = (float*)d_ws;                          // [2048][69][384]
  float* attn_ws = qkv_ws + (size_t)B_TOT * NTOK * QKV3;  // [2048][69][128]
  float* pp_ws   = attn_ws + (size_t)B_TOT * NTOK * CDIM; // [2048][20][128]

  qkv_kernel<<<B_TOT, 256, 0, stream>>>(x, spa, Wqkv, bqkv, qkv_ws);
  attn_kernel<<<B_TOT * NH, 256, 0, stream>>>(qkv_ws, relb, mask, attnw_out, attn_ws);
  proj_kernel<<<B_TOT, 256, 0, stream>>>(attn_ws, Wproj, bproj, x_out, pp_ws);
  prompt_reduce_kernel<<<(PROMPT_ELEMS + 255) / 256, 256, 0, stream>>>(pp_ws, prompts_out);
}